// GraphModel_67138928771353
// MI455X (gfx1250) — compile-verified
//
#include <hip/hip_runtime.h>
#include <math.h>

typedef __attribute__((ext_vector_type(2))) float v2f;
typedef __attribute__((ext_vector_type(8))) float v8f;

#define NEG_SLOPE 0.2f

// ---------------------------------------------------------------------------
// float atomic max via int/uint punning (monotone bit orderings, no NaNs here)
// ---------------------------------------------------------------------------
__device__ inline void atomicMaxF(float* addr, float val) {
    if (val >= 0.0f)
        atomicMax((int*)addr, __float_as_int(val));
    else
        atomicMin((unsigned int*)addr, __float_as_uint(val));
}

// ---------------------------------------------------------------------------
// fill
// ---------------------------------------------------------------------------
__global__ void fill_kernel(float* __restrict__ p, float v, int n) {
    int i = blockIdx.x * blockDim.x + threadIdx.x;
    if (i < n) p[i] = v;
}

// ---------------------------------------------------------------------------
// out[M x NC] = A[M x K] * W[K x NC] + bias  (f32, V_WMMA_F32_16X16X4_F32)
// One wave (32 threads) per 16x64 tile: 4 accumulators share each A fragment.
// A fragment layout (16x4 f32): lanes 0-15 row M=lane, v0=K0 v1=K1;
//                               lanes 16-31 same rows, v0=K2 v1=K3.
// B fragment (4x16): lanes 0-15 col N=lane, v0=K0 v1=K1; lanes 16-31 K2/K3.
// C/D: VGPR r -> row r (lanes 0-15) / row r+8 (lanes 16-31), col = lane&15.
// Requires: M%16==0, K%4==0, NC%64==0 (true here: M=50000, K in {128,64},
// NC in {256,512}). All lanes active -> EXEC all-ones around WMMA.
// ---------------------------------------------------------------------------
__global__ void gemm_bias_wmma(const float* __restrict__ A,
                               const float* __restrict__ W,
                               const float* __restrict__ bias,
                               float* __restrict__ out,
                               int K, int NC) {
    int lane = threadIdx.x & 31;
    int m0 = blockIdx.x * 16;
    int n0 = blockIdx.y * 64;
    int rowA = m0 + (lane & 15);
    int col  = n0 + (lane & 15);
    int koff = (lane >> 4) << 1;              // 0 or 2

    const float* Ap = A + (size_t)rowA * K + koff;
    v8f acc0 = {}, acc1 = {}, acc2 = {}, acc3 = {};

    for (int k = 0; k < K; k += 4) {
        v2f a;
        a.x = Ap[k];
        a.y = Ap[k + 1];
        const float* W0 = W + (size_t)(k + koff) * NC + col;
        const float* W1 = W0 + NC;
        v2f b0, b1, b2, b3;
        b0.x = W0[0];  b0.y = W1[0];
        b1.x = W0[16]; b1.y = W1[16];
        b2.x = W0[32]; b2.y = W1[32];
        b3.x = W0[48]; b3.y = W1[48];
        acc0 = __builtin_amdgcn_wmma_f32_16x16x4_f32(false, a, false, b0, (short)0, acc0, false, false);
        acc1 = __builtin_amdgcn_wmma_f32_16x16x4_f32(false, a, false, b1, (short)0, acc1, false, false);
        acc2 = __builtin_amdgcn_wmma_f32_16x16x4_f32(false, a, false, b2, (short)0, acc2, false, false);
        acc3 = __builtin_amdgcn_wmma_f32_16x16x4_f32(false, a, false, b3, (short)0, acc3, false, false);
    }

    float bv0 = bias[col];
    float bv1 = bias[col + 16];
    float bv2 = bias[col + 32];
    float bv3 = bias[col + 48];
    int mAdd = (lane >> 4) << 3;              // 0 or 8
#pragma unroll
    for (int r = 0; r < 8; ++r) {
        size_t o = (size_t)(m0 + mAdd + r) * NC + col;
        out[o]      = acc0[r] + bv0;
        out[o + 16] = acc1[r] + bv1;
        out[o + 32] = acc2[r] + bv2;
        out[o + 48] = acc3[r] + bv3;
    }
}

// ---------------------------------------------------------------------------
// Per-edge attention logits: logit[e,h] = attn[h,:] . leakyrelu(fs[src]+fd[dst])
// One wave per edge; cross-lane reduction; atomic segment max over dst.
// ---------------------------------------------------------------------------
__global__ void edge_logits_kernel(const float* __restrict__ fs,
                                   const float* __restrict__ fd,
                                   const float* __restrict__ attn,
                                   const int* __restrict__ src,
                                   const int* __restrict__ dst,
                                   float* __restrict__ logits,
                                   float* __restrict__ segmax,
                                   int E, int D) {
    int wid = (blockIdx.x * blockDim.x + threadIdx.x) >> 5;
    int lane = threadIdx.x & 31;
    if (wid >= E) return;
    int s = src[wid], d = dst[wid];
    int ld = 4 * D;
    const float* ps = fs + (size_t)s * ld;
    const float* pd = fd + (size_t)d * ld;
#pragma unroll
    for (int h = 0; h < 4; ++h) {
        float a = 0.0f;
        for (int i = lane; i < D; i += 32) {
            int idx = h * D + i;
            float v = ps[idx] + pd[idx];
            v = (v > 0.0f) ? v : NEG_SLOPE * v;
            a += v * attn[idx];
        }
#pragma unroll
        for (int off = 16; off > 0; off >>= 1) a += __shfl_xor(a, off, 32);
        if (lane == 0) {
            logits[(size_t)wid * 4 + h] = a;
            atomicMaxF(&segmax[d * 4 + h], a);
        }
    }
}

// ---------------------------------------------------------------------------
// ex = exp(logit - segmax[dst]); store in place; atomic segment sum.
// ---------------------------------------------------------------------------
__global__ void edge_exp_kernel(float* __restrict__ logits,
                                const float* __restrict__ segmax,
                                float* __restrict__ segsum,
                                const int* __restrict__ dst, int E) {
    int i = blockIdx.x * blockDim.x + threadIdx.x;
    if (i >= E * 4) return;
    int e = i >> 2, h = i & 3;
    int d = dst[e];
    float ex = expf(logits[i] - segmax[d * 4 + h]);
    logits[i] = ex;
    atomicAdd(&segsum[d * 4 + h], ex);
}

// ---------------------------------------------------------------------------
// agg[dst,h,:] += (ex/segsum) * fs[src,h,:]   (one wave per edge)
// ---------------------------------------------------------------------------
__global__ void edge_agg_kernel(const float* __restrict__ fs,
                                const float* __restrict__ ex,
                                const float* __restrict__ segsum,
                                const int* __restrict__ src,
                                const int* __restrict__ dst,
                                float* __restrict__ agg,
                                int E, int D) {
    int wid = (blockIdx.x * blockDim.x + threadIdx.x) >> 5;
    int lane = threadIdx.x & 31;
    if (wid >= E) return;
    int s = src[wid], d = dst[wid];
    float a0 = ex[(size_t)wid * 4 + 0] / segsum[d * 4 + 0];
    float a1 = ex[(size_t)wid * 4 + 1] / segsum[d * 4 + 1];
    float a2 = ex[(size_t)wid * 4 + 2] / segsum[d * 4 + 2];
    float a3 = ex[(size_t)wid * 4 + 3] / segsum[d * 4 + 3];
    int ld = 4 * D;
    const float* ps = fs + (size_t)s * ld;
    float* pa = agg + (size_t)d * ld;
    for (int i = lane; i < ld; i += 32) {
        int h = i / D;
        float al = (h == 0) ? a0 : (h == 1) ? a1 : (h == 2) ? a2 : a3;
        atomicAdd(&pa[i], al * ps[i]);
    }
}

// ---------------------------------------------------------------------------
// head-wise max pool: out[n,d] = max_h agg[n,h,d]
// ---------------------------------------------------------------------------
__global__ void head_maxpool_kernel(const float* __restrict__ agg,
                                    float* __restrict__ out, int N, int D) {
    int i = blockIdx.x * blockDim.x + threadIdx.x;
    if (i >= N * D) return;
    int n = i / D, dd = i % D;
    const float* p = agg + (size_t)n * 4 * D + dd;
    float m = p[0];
    m = fmaxf(m, p[D]);
    m = fmaxf(m, p[2 * D]);
    m = fmaxf(m, p[3 * D]);
    out[i] = m;
}

// ---------------------------------------------------------------------------
// gate[n] = h2[n,:] . gate_w + gate_b ; atomic per-graph max  (wave per node)
// ---------------------------------------------------------------------------
__global__ void gate_kernel(const float* __restrict__ h2,
                            const float* __restrict__ gw,
                            const float* __restrict__ gb,
                            const int* __restrict__ n2g,
                            float* __restrict__ gate,
                            float* __restrict__ gmax, int N) {
    int n = (blockIdx.x * blockDim.x + threadIdx.x) >> 5;
    int lane = threadIdx.x & 31;
    if (n >= N) return;
    const float* p = h2 + (size_t)n * 128;
    float a = 0.0f;
    for (int i = lane; i < 128; i += 32) a += p[i] * gw[i];
#pragma unroll
    for (int off = 16; off > 0; off >>= 1) a += __shfl_xor(a, off, 32);
    if (lane == 0) {
        a += gb[0];
        gate[n] = a;
        atomicMaxF(&gmax[n2g[n]], a);
    }
}

__global__ void gate_exp_kernel(float* __restrict__ gate,
                                const float* __restrict__ gmax,
                                float* __restrict__ gsum,
                                const int* __restrict__ n2g, int N) {
    int n = blockIdx.x * blockDim.x + threadIdx.x;
    if (n >= N) return;
    float ge = expf(gate[n] - gmax[n2g[n]]);
    gate[n] = ge;
    atomicAdd(&gsum[n2g[n]], ge);
}

__global__ void graph_pool_kernel(const float* __restrict__ h2,
                                  const float* __restrict__ gate,
                                  const float* __restrict__ gsum,
                                  const int* __restrict__ n2g,
                                  float* __restrict__ out, int N) {
    int i = blockIdx.x * blockDim.x + threadIdx.x;
    if (i >= N * 128) return;
    int n = i >> 7, dd = i & 127;
    int g = n2g[n];
    atomicAdd(&out[g * 128 + dd], (gate[n] / gsum[g]) * h2[i]);
}

// ---------------------------------------------------------------------------
// launch
// ---------------------------------------------------------------------------
extern "C" void kernel_launch(void* const* d_in, const int* in_sizes, int n_in,
                              void* d_out, int out_size, void* d_ws, size_t ws_size,
                              hipStream_t stream) {
    const float* x     = (const float*)d_in[0];
    const int*   src   = (const int*)d_in[1];
    const int*   dst   = (const int*)d_in[2];
    const int*   n2g   = (const int*)d_in[3];
    const float* Ws1   = (const float*)d_in[4];
    const float* bs1   = (const float*)d_in[5];
    const float* Wd1   = (const float*)d_in[6];
    const float* bd1   = (const float*)d_in[7];
    const float* attn1 = (const float*)d_in[8];
    const float* Ws2   = (const float*)d_in[9];
    const float* bs2   = (const float*)d_in[10];
    const float* Wd2   = (const float*)d_in[11];
    const float* bd2   = (const float*)d_in[12];
    const float* attn2 = (const float*)d_in[13];
    const float* gw    = (const float*)d_in[14];
    const float* gb    = (const float*)d_in[15];
    float* out = (float*)d_out;

    const int N = in_sizes[0] / 128;   // 50000
    const int E = in_sizes[1];         // 320000
    const int G = out_size / 128;      // 50

    // workspace carve-up (floats)
    float* fS   = (float*)d_ws;                 // N*512 : fs1 (first N*256) / fs2
    float* fD   = fS   + (size_t)N * 512;       // N*512 : fd1/fd2, reused as agg
    float* hb   = fD   + (size_t)N * 512;       // N*128 : h1 (first N*64) then h2
    float* lg   = hb   + (size_t)N * 128;       // E*4   : logits -> exp in place
    float* smax = lg   + (size_t)E * 4;         // N*4
    float* ssum = smax + (size_t)N * 4;         // N*4
    float* gate = ssum + (size_t)N * 4;         // N
    float* gmax = gate + N;                     // G
    float* gsum = gmax + G;                     // G

    const int TB = 256;
    dim3 blk(TB);
    int ewBlocks = (E + 7) / 8;                 // wave-per-edge, 8 waves/block

    // ---------------- layer 1 (IN=128 -> HEADS*64) ----------------
    fill_kernel<<<(N * 4 + TB - 1) / TB, blk, 0, stream>>>(smax, -INFINITY, N * 4);
    fill_kernel<<<(N * 4 + TB - 1) / TB, blk, 0, stream>>>(ssum, 0.0f, N * 4);
    gemm_bias_wmma<<<dim3(N / 16, 4), dim3(32), 0, stream>>>(x, Ws1, bs1, fS, 128, 256);
    gemm_bias_wmma<<<dim3(N / 16, 4), dim3(32), 0, stream>>>(x, Wd1, bd1, fD, 128, 256);
    edge_logits_kernel<<<ewBlocks, blk, 0, stream>>>(fS, fD, attn1, src, dst, lg, smax, E, 64);
    edge_exp_kernel<<<(E * 4 + TB - 1) / TB, blk, 0, stream>>>(lg, smax, ssum, dst, E);
    fill_kernel<<<(N * 256 + TB - 1) / TB, blk, 0, stream>>>(fD, 0.0f, N * 256);
    edge_agg_kernel<<<ewBlocks, blk, 0, stream>>>(fS, lg, ssum, src, dst, fD, E, 64);
    head_maxpool_kernel<<<(N * 64 + TB - 1) / TB, blk, 0, stream>>>(fD, hb, N, 64);

    // ---------------- layer 2 (64 -> HEADS*128) ----------------
    fill_kernel<<<(N * 4 + TB - 1) / TB, blk, 0, stream>>>(smax, -INFINITY, N * 4);
    fill_kernel<<<(N * 4 + TB - 1) / TB, blk, 0, stream>>>(ssum, 0.0f, N * 4);
    gemm_bias_wmma<<<dim3(N / 16, 8), dim3(32), 0, stream>>>(hb, Ws2, bs2, fS, 64, 512);
    gemm_bias_wmma<<<dim3(N / 16, 8), dim3(32), 0, stream>>>(hb, Wd2, bd2, fD, 64, 512);
    edge_logits_kernel<<<ewBlocks, blk, 0, stream>>>(fS, fD, attn2, src, dst, lg, smax, E, 128);
    edge_exp_kernel<<<(E * 4 + TB - 1) / TB, blk, 0, stream>>>(lg, smax, ssum, dst, E);
    fill_kernel<<<(N * 512 + TB - 1) / TB, blk, 0, stream>>>(fD, 0.0f, N * 512);
    edge_agg_kernel<<<ewBlocks, blk, 0, stream>>>(fS, lg, ssum, src, dst, fD, E, 128);
    head_maxpool_kernel<<<(N * 128 + TB - 1) / TB, blk, 0, stream>>>(fD, hb, N, 128);

    // ---------------- global attention pooling ----------------
    fill_kernel<<<1, 64, 0, stream>>>(gmax, -INFINITY, G);
    fill_kernel<<<1, 64, 0, stream>>>(gsum, 0.0f, G);
    fill_kernel<<<(G * 128 + TB - 1) / TB, blk, 0, stream>>>(out, 0.0f, G * 128);
    gate_kernel<<<(N * 32 + TB - 1) / TB, blk, 0, stream>>>(hb, gw, gb, n2g, gate, gmax, N);
    gate_exp_kernel<<<(N + TB - 1) / TB, blk, 0, stream>>>(gate, gmax, gsum, n2g, N);
    graph_pool_kernel<<<(N * 128 + TB - 1) / TB, blk, 0, stream>>>(hb, gate, gsum, n2g, out, N);
}